// HeteroClassifier_90271622627983
// MI455X (gfx1250) — compile-verified
//
#include <hip/hip_runtime.h>
#include <math.h>

// ---------------- problem constants (match reference) ----------------
#define B_G   1024
#define NQn   16384
#define NWn   65536
#define NIn   8192
#define EWQn  131072
#define EIQn  32768
#define EQQn  65536
#define VOC   30000
#define DIN   50
#define DH    32
#define DIMG  512
#define EPSc  1e-7f

typedef float v2f __attribute__((ext_vector_type(2)));
typedef float v8f __attribute__((ext_vector_type(8)));

// ---------------- utility kernels ----------------
__global__ void zero_f32(float* __restrict__ p, long long n) {
  long long i = (long long)blockIdx.x * blockDim.x + threadIdx.x;
  long long s = (long long)gridDim.x * blockDim.x;
  for (; i < n; i += s) p[i] = 0.0f;
}

__global__ void degree_accum(const int* __restrict__ src, const int* __restrict__ dst,
                             float* __restrict__ deg_o, float* __restrict__ deg_i, int E) {
  int e = blockIdx.x * blockDim.x + threadIdx.x;
  if (e < E) {
    atomicAdd(&deg_o[src[e]], 1.0f);
    atomicAdd(&deg_i[dst[e]], 1.0f);
  }
}

__global__ void count_accum(const int* __restrict__ gid, float* __restrict__ cnt, int n) {
  int q = blockIdx.x * blockDim.x + threadIdx.x;
  if (q < n) atomicAdd(&cnt[gid[q]], 1.0f);
}

// p[i] = (max(p[i],1))^-1/2
__global__ void finalize_dinv(float* __restrict__ p, long long n) {
  long long i = (long long)blockIdx.x * blockDim.x + threadIdx.x;
  long long s = (long long)gridDim.x * blockDim.x;
  for (; i < n; i += s) p[i] = 1.0f / sqrtf(fmaxf(p[i], 1.0f));
}

// ---------------- f32 WMMA GEMM: C = A[MxK] * B[KxN] (+bias) ----------------
// Requires M % 16 == 0 and N % 16 == 0 (true at every call site); K arbitrary.
// 4 waves per block; one wave per 16x16 C tile. Main K loop is branch-free
// (unconditional b64 A loads + b32 B loads); K%4 tail uses clamped addresses
// + selects so no exec-mask churn around the V_WMMA_F32_16X16X4_F32 chain.
__global__ void wmma_gemm_f32(const float* __restrict__ A, const float* __restrict__ Bm,
                              float* __restrict__ C, const float* __restrict__ bias,
                              int M, int N, int K, int lda, int ldb, int ldc, int relu) {
  const int wave  = threadIdx.x >> 5;   // 0..3
  const int lane  = threadIdx.x & 31;
  const int tileM = (blockIdx.x * 4 + wave) * 16;
  const int tileN = blockIdx.y * 16;
  if (tileM >= M) return;               // wave-uniform: EXEC stays all-ones

  const int half = lane >> 4;           // 0: K pair {0,1}, 1: K pair {2,3}
  const int l15  = lane & 15;

  const float* Arow = A + (long long)(tileM + l15) * lda + half * 2;  // 8B aligned
  const float* Bcol = Bm + tileN + l15 + (long long)(half * 2) * ldb;

  v8f acc = {};
  const int Kmain = K & ~3;
  for (int k0 = 0; k0 < Kmain; k0 += 4) {
    v2f a = *(const v2f*)(Arow + k0);          // A[row][k0+koff .. +1] contiguous
    v2f b;
    b.x = Bcol[(long long)k0 * ldb];
    b.y = Bcol[(long long)(k0 + 1) * ldb];
    acc = __builtin_amdgcn_wmma_f32_16x16x4_f32(false, a, false, b, (short)0, acc,
                                                false, false);
  }
  if (Kmain < K) {                             // K % 4 tail, branch-free selects
    const float* Abase = A + (long long)(tileM + l15) * lda;
    const int ka = Kmain + half * 2, kb = ka + 1;
    const int kac = ka < K ? ka : K - 1;
    const int kbc = kb < K ? kb : K - 1;
    v2f a, b;
    a.x = Abase[kac];                        a.x = ka < K ? a.x : 0.0f;
    a.y = Abase[kbc];                        a.y = kb < K ? a.y : 0.0f;
    b.x = Bm[(long long)kac * ldb + tileN + l15]; b.x = ka < K ? b.x : 0.0f;
    b.y = Bm[(long long)kbc * ldb + tileN + l15]; b.y = kb < K ? b.y : 0.0f;
    acc = __builtin_amdgcn_wmma_f32_16x16x4_f32(false, a, false, b, (short)0, acc,
                                                false, false);
  }

  // C layout: VGPR r -> row tileM + r + half*8, col tileN + l15
  const int ccol = tileN + l15;
  const float bv = bias ? bias[ccol] : 0.0f;
  float* Cb = C + (long long)(tileM + half * 8) * ldc + ccol;
#pragma unroll
  for (int r = 0; r < 8; ++r) {
    float v = acc[r] + bv;
    if (relu) v = fmaxf(v, 0.0f);
    Cb[(long long)r * ldc] = v;
  }
}

// btS1[d] = sum_k bt[k] * Ws1[k][d]
__global__ void small_vecmat(const float* __restrict__ bt, const float* __restrict__ Ws1,
                             float* __restrict__ out) {
  int d = threadIdx.x;  // 32 threads
  float s = 0.0f;
  for (int k = 0; k < DIN; ++k) s += bt[k] * Ws1[k * DH + d];
  out[d] = s;
}

// ---------------- scatter (message passing) ----------------
// word->query: msg = WeS1[word_id[src]] * dinv_o[src]; one lane per (edge,dim)
__global__ void scatter_word(const int* __restrict__ src, const int* __restrict__ dst,
                             const int* __restrict__ word_id,
                             const float* __restrict__ WeS1,
                             const float* __restrict__ dinv_o,
                             float* __restrict__ agg, int E) {
  int tid = blockIdx.x * blockDim.x + threadIdx.x;
  int e = tid >> 5, d = tid & 31;
  if (e >= E) return;
  int s = src[e], t = dst[e];
  float v = WeS1[(long long)word_id[s] * DH + d] * dinv_o[s];
  atomicAdd(&agg[(long long)t * DH + d], v);
}

// generic feature scatter: msg = feat[src] * dinv_o[src]
__global__ void scatter_feat(const int* __restrict__ src, const int* __restrict__ dst,
                             const float* __restrict__ feat,
                             const float* __restrict__ dinv_o,
                             float* __restrict__ agg, int E) {
  int tid = blockIdx.x * blockDim.x + threadIdx.x;
  int e = tid >> 5, d = tid & 31;
  if (e >= E) return;
  int s = src[e], t = dst[e];
  float v = feat[(long long)s * DH + d] * dinv_o[s];
  atomicAdd(&agg[(long long)t * DH + d], v);
}

// h1 = relu(agg_wq*dinv_i_wq + agg_iq*dinv_i_iq + 2*bs1 + bu1)
__global__ void h1_kernel(const float* __restrict__ agg_wq, const float* __restrict__ agg_iq,
                          const float* __restrict__ dinv_i_wq, const float* __restrict__ dinv_i_iq,
                          const float* __restrict__ bs1, const float* __restrict__ bu1,
                          float* __restrict__ h1) {
  int tid = blockIdx.x * blockDim.x + threadIdx.x;
  int q = tid >> 5, d = tid & 31;
  if (q >= NQn) return;
  float v = agg_wq[tid] * dinv_i_wq[q] + agg_iq[tid] * dinv_i_iq[q] + 2.0f * bs1[d] + bu1[d];
  h1[tid] = fmaxf(v, 0.0f);
}

// h2 = agg_qq*dinv_i_qq + bu2 ; scatter into per-graph sum
__global__ void pool_kernel(const float* __restrict__ agg_qq, const float* __restrict__ dinv_i_qq,
                            const float* __restrict__ bu2, const int* __restrict__ qgid,
                            float* __restrict__ emb) {
  int tid = blockIdx.x * blockDim.x + threadIdx.x;
  int q = tid >> 5, d = tid & 31;
  if (q >= NQn) return;
  float v = agg_qq[tid] * dinv_i_qq[q] + bu2[d];
  atomicAdd(&emb[(long long)qgid[q] * DH + d], v);
}

__global__ void mean_kernel(float* __restrict__ emb, const float* __restrict__ cnt) {
  int tid = blockIdx.x * blockDim.x + threadIdx.x;
  int g = tid >> 5;
  if (g >= B_G) return;
  emb[tid] /= fmaxf(cnt[g], 1.0f);
}

// ---------------- loss kernels ----------------
// s1: one wave per sample; hidden = relu(ua[i]+vp[i]+bd0); s = sigmoid(dot(h,Wd1)+bd1)
__global__ void s1_loss(const float* __restrict__ ua, const float* __restrict__ vp,
                        const float* __restrict__ bd0, const float* __restrict__ Wd1,
                        const float* __restrict__ bd1, float* __restrict__ loss) {
  int i = blockIdx.x, d = threadIdx.x;  // 32 threads
  float h = fmaxf(ua[i * DH + d] + vp[i * DH + d] + bd0[d], 0.0f) * Wd1[d];
#pragma unroll
  for (int off = 16; off > 0; off >>= 1) h += __shfl_xor(h, off, 32);
  if (d == 0) {
    float s = 1.0f / (1.0f + expf(-(h + bd1[0])));
    s = fminf(fmaxf(s, EPSc), 1.0f);
    atomicAdd(loss, -logf(s) * (1.0f / (float)B_G));
  }
}

// s2: one block per row i; max_j sigmoid(dot(relu(ua[i]+va[j]+bd0),Wd1)+bd1)*mask[i,j]
__global__ void s2_loss(const float* __restrict__ ua, const float* __restrict__ va,
                        const float* __restrict__ bd0, const float* __restrict__ Wd1,
                        const float* __restrict__ bd1, const int* __restrict__ mask,
                        float* __restrict__ loss) {
  __shared__ float t[DH], w[DH];
  __shared__ float red[256];
  int i = blockIdx.x;
  if (threadIdx.x < DH) {
    t[threadIdx.x] = ua[i * DH + threadIdx.x] + bd0[threadIdx.x];
    w[threadIdx.x] = Wd1[threadIdx.x];
  }
  __syncthreads();
  const float bb = bd1[0];
  float m = 0.0f;  // all row entries are >= 0 (sigmoid>0, mask in {0,1})
  for (int j = threadIdx.x; j < B_G; j += blockDim.x) {
    const float* vj = va + (long long)j * DH;
    float acc = 0.0f;
#pragma unroll
    for (int d = 0; d < DH; ++d) acc += fmaxf(t[d] + vj[d], 0.0f) * w[d];
    float s = 1.0f / (1.0f + expf(-(acc + bb)));
    s *= (float)mask[(long long)i * B_G + j];
    m = fmaxf(m, s);
  }
  red[threadIdx.x] = m;
  __syncthreads();
  for (int off = 128; off > 0; off >>= 1) {
    if (threadIdx.x < off) red[threadIdx.x] = fmaxf(red[threadIdx.x], red[threadIdx.x + off]);
    __syncthreads();
  }
  if (threadIdx.x == 0) {
    float v = fminf(fmaxf(1.0f - red[0], EPSc), 1.0f);
    atomicAdd(loss, -logf(v) * (1.0f / (float)B_G));
  }
}

// ---------------- host orchestration ----------------
extern "C" void kernel_launch(void* const* d_in, const int* in_sizes, int n_in,
                              void* d_out, int out_size, void* d_ws, size_t ws_size,
                              hipStream_t stream) {
  (void)in_sizes; (void)n_in; (void)ws_size;
  const int*   word_id_a = (const int*)d_in[0];
  const int*   word_id_p = (const int*)d_in[1];
  const float* img_a     = (const float*)d_in[2];
  const float* img_p     = (const float*)d_in[3];
  const int*   src_wq    = (const int*)d_in[4];
  const int*   dst_wq    = (const int*)d_in[5];
  const int*   src_iq    = (const int*)d_in[6];
  const int*   dst_iq    = (const int*)d_in[7];
  const int*   src_qq    = (const int*)d_in[8];
  const int*   dst_qq    = (const int*)d_in[9];
  const int*   qgid      = (const int*)d_in[10];
  const int*   click     = (const int*)d_in[11];
  const float* wordemb   = (const float*)d_in[12];
  const float* Wt        = (const float*)d_in[13];
  const float* bt        = (const float*)d_in[14];
  const float* Ws1       = (const float*)d_in[15];
  const float* bs1       = (const float*)d_in[16];
  /* Wu1 (d_in[17]) multiplies zero features -> unused */
  const float* bu1       = (const float*)d_in[18];
  const float* Wu2       = (const float*)d_in[19];
  const float* bu2       = (const float*)d_in[20];
  const float* Wd0       = (const float*)d_in[21];
  const float* bd0       = (const float*)d_in[22];
  const float* Wd1       = (const float*)d_in[23];
  const float* bd1       = (const float*)d_in[24];
  float* loss = (float*)d_out;
  float* ws = (float*)d_ws;

  // ---- workspace layout (floats) ----
  size_t o = 0;
  float* dinv_o_wq = ws + o; o += NWn;   // degree arrays (shared across sides)
  float* dinv_i_wq = ws + o; o += NQn;
  float* dinv_o_iq = ws + o; o += NIn;
  float* dinv_i_iq = ws + o; o += NQn;
  float* dinv_o_qq = ws + o; o += NQn;
  float* dinv_i_qq = ws + o; o += NQn;
  const size_t dinv_total = o;
  float* cnt   = ws + o; o += B_G;
  float* WeS1  = ws + o; o += (size_t)VOC * DH;   // wordemb @ Ws1
  float* WtS1  = ws + o; o += (size_t)DIMG * DH;  // Wt @ Ws1
  float* btS1  = ws + o; o += DH;                 // bt @ Ws1
  float* emb_a = ws + o; o += (size_t)B_G * DH;
  float* emb_p = ws + o; o += (size_t)B_G * DH;   // adjacent to emb_a (zeroed together)
  float* ua    = ws + o; o += (size_t)B_G * DH;
  float* va    = ws + o; o += (size_t)B_G * DH;
  float* vp    = ws + o; o += (size_t)B_G * DH;
  // per-side scratch (reused): three adjacent agg buffers first
  float* agg_wq = ws + o; o += (size_t)NQn * DH;
  float* agg_iq = ws + o; o += (size_t)NQn * DH;
  float* agg_qq = ws + o; o += (size_t)NQn * DH;
  float* mi     = ws + o; o += (size_t)NIn * DH;
  float* h1     = ws + o; o += (size_t)NQn * DH;
  float* m2     = ws + o; o += (size_t)NQn * DH;

  const int TB = 256;
  auto nb  = [](long long n, int tb) { return (unsigned)((n + tb - 1) / tb); };
  auto gx  = [](int M) { return (unsigned)((M / 16 + 3) / 4); };  // 4 tiles/block

  // ---- zero accumulators: loss, degrees+cnt, emb_a/emb_p ----
  zero_f32<<<1, 32, 0, stream>>>(loss, out_size);
  zero_f32<<<512, TB, 0, stream>>>(dinv_o_wq, (long long)(dinv_total + B_G));
  zero_f32<<<256, TB, 0, stream>>>(emb_a, (long long)2 * B_G * DH);

  // ---- degrees (shared, edges identical for both sides) ----
  degree_accum<<<nb(EWQn, TB), TB, 0, stream>>>(src_wq, dst_wq, dinv_o_wq, dinv_i_wq, EWQn);
  degree_accum<<<nb(EIQn, TB), TB, 0, stream>>>(src_iq, dst_iq, dinv_o_iq, dinv_i_iq, EIQn);
  degree_accum<<<nb(EQQn, TB), TB, 0, stream>>>(src_qq, dst_qq, dinv_o_qq, dinv_i_qq, EQQn);
  count_accum<<<nb(NQn, TB), TB, 0, stream>>>(qgid, cnt, NQn);
  finalize_dinv<<<512, TB, 0, stream>>>(dinv_o_wq, (long long)dinv_total);

  // ---- fold linears (shared): WeS1 = wordemb@Ws1, WtS1 = Wt@Ws1, btS1 = bt@Ws1 ----
  wmma_gemm_f32<<<dim3(gx(VOC), DH / 16), 128, 0, stream>>>(
      wordemb, Ws1, WeS1, nullptr, VOC, DH, DIN, DIN, DH, DH, 0);
  wmma_gemm_f32<<<dim3(gx(DIMG), DH / 16), 128, 0, stream>>>(
      Wt, Ws1, WtS1, nullptr, DIMG, DH, DIN, DIN, DH, DH, 0);
  small_vecmat<<<1, 32, 0, stream>>>(bt, Ws1, btS1);

  // ---- per-side graph embedding (sequential, reusing scratch) ----
  for (int side = 0; side < 2; ++side) {
    const int*   word_id = side == 0 ? word_id_a : word_id_p;
    const float* img     = side == 0 ? img_a : img_p;
    float*       emb     = side == 0 ? emb_a : emb_p;

    zero_f32<<<512, TB, 0, stream>>>(agg_wq, (long long)3 * NQn * DH);
    // mi = img @ (Wt@Ws1) + bt@Ws1   [8192 x 32], K = 512  -> WMMA
    wmma_gemm_f32<<<dim3(gx(NIn), DH / 16), 128, 0, stream>>>(
        img, WtS1, mi, btS1, NIn, DH, DIMG, DIMG, DH, DH, 0);
    // message passing (norm='both')
    scatter_word<<<nb((long long)EWQn * 32, TB), TB, 0, stream>>>(
        src_wq, dst_wq, word_id, WeS1, dinv_o_wq, agg_wq, EWQn);
    scatter_feat<<<nb((long long)EIQn * 32, TB), TB, 0, stream>>>(
        src_iq, dst_iq, mi, dinv_o_iq, agg_iq, EIQn);
    h1_kernel<<<nb((long long)NQn * 32, TB), TB, 0, stream>>>(
        agg_wq, agg_iq, dinv_i_wq, dinv_i_iq, bs1, bu1, h1);
    // m2 = h1 @ Wu2   [16384 x 32], K = 32 -> WMMA
    wmma_gemm_f32<<<dim3(gx(NQn), DH / 16), 128, 0, stream>>>(
        h1, Wu2, m2, nullptr, NQn, DH, DH, DH, DH, DH, 0);
    scatter_feat<<<nb((long long)EQQn * 32, TB), TB, 0, stream>>>(
        src_qq, dst_qq, m2, dinv_o_qq, agg_qq, EQQn);
    pool_kernel<<<nb((long long)NQn * 32, TB), TB, 0, stream>>>(
        agg_qq, dinv_i_qq, bu2, qgid, emb);
    mean_kernel<<<nb((long long)B_G * 32, TB), TB, 0, stream>>>(emb, cnt);
  }

  // ---- discriminator, factored: concat(x1,x2)@Wd0 = x1@Wd0_top + x2@Wd0_bot ----
  const float* Wd0_top = Wd0;
  const float* Wd0_bot = Wd0 + (size_t)DH * DH;  // rows 32..63
  wmma_gemm_f32<<<dim3(gx(B_G), DH / 16), 128, 0, stream>>>(
      emb_a, Wd0_top, ua, nullptr, B_G, DH, DH, DH, DH, DH, 0);
  wmma_gemm_f32<<<dim3(gx(B_G), DH / 16), 128, 0, stream>>>(
      emb_a, Wd0_bot, va, nullptr, B_G, DH, DH, DH, DH, DH, 0);
  wmma_gemm_f32<<<dim3(gx(B_G), DH / 16), 128, 0, stream>>>(
      emb_p, Wd0_bot, vp, nullptr, B_G, DH, DH, DH, DH, DH, 0);

  // ---- losses ----
  s1_loss<<<B_G, 32, 0, stream>>>(ua, vp, bd0, Wd1, bd1, loss);
  s2_loss<<<B_G, 256, 0, stream>>>(ua, va, bd0, Wd1, bd1, click, loss);
}